// Net_18794776887753
// MI455X (gfx1250) — compile-verified
//
#include <hip/hip_runtime.h>

typedef float v2f __attribute__((ext_vector_type(2)));
typedef float v8f __attribute__((ext_vector_type(8)));

#define NN 100000

__device__ __forceinline__ void atomAddF(float* p, float v) {
  unsafeAtomicAdd(p, v);   // lowers to global_atomic_add_f32 on gfx1250
}

// ---------------- degree / dinv ----------------
__global__ void k_fill(float* __restrict__ p, float v, int n) {
  int i = blockIdx.x * blockDim.x + threadIdx.x;
  if (i < n) p[i] = v;
}

__global__ void k_deg_accum(float* __restrict__ deg, const int* __restrict__ cols, int E) {
  int i = blockIdx.x * blockDim.x + threadIdx.x;
  if (i < E) atomAddF(&deg[cols[i]], 1.0f);
}

__global__ void k_rsqrt_inplace(float* __restrict__ p, int n) {
  int i = blockIdx.x * blockDim.x + threadIdx.x;
  if (i < n) {
    float d = p[i];
    p[i] = (d > 0.0f) ? rsqrtf(d) : 0.0f;
  }
}

// ---------------- fp32 WMMA GEMM ----------------
// One wave computes a 16-row strip x N columns (NT tiles of 16).
// A frag (16x4 f32): lane = M (lo 16 lanes K=0,1; hi lanes K=2,3), 2 VGPRs.
// B frag (4x16 f32): lane = N (lo lanes K=0,1; hi lanes K=2,3), 2 VGPRs.
// C/D (16x16 f32): VGPR j -> row j (lanes 0-15) / row j+8 (lanes 16-31), lane = col.
__device__ __forceinline__ v8f wmma_f32x4(v2f a, v2f b, v8f c) {
  return __builtin_amdgcn_wmma_f32_16x16x4_f32(false, a, false, b, (short)0, c, false, false);
}

template <int K, int N, int NT>
__global__ void k_gemm_wmma(const float* __restrict__ X, const float* __restrict__ W,
                            float* __restrict__ H, int strips) {
  int wave = (int)((blockIdx.x * blockDim.x + threadIdx.x) >> 5);
  if (wave >= strips) return;           // wave-uniform: EXEC stays all-ones
  int lane = threadIdx.x & 31;
  int m  = lane & 15;                   // M for A, N-within-tile for B/D
  int kh = lane >> 4;                   // K-half select

  const float* xrow = X + (size_t)(wave * 16 + m) * K + 2 * kh;

  v8f acc[NT];
#pragma unroll
  for (int t = 0; t < NT; ++t) acc[t] = (v8f){0.f, 0.f, 0.f, 0.f, 0.f, 0.f, 0.f, 0.f};

  for (int k = 0; k < K; k += 4) {
    v2f a = *(const v2f*)(xrow + k);
    const float* wk = W + (size_t)(k + 2 * kh) * N + m;
#pragma unroll
    for (int t = 0; t < NT; ++t) {
      v2f b;
      b.x = wk[t * 16];
      b.y = wk[t * 16 + N];
      acc[t] = wmma_f32x4(a, b, acc[t]);
    }
  }

  int orow = wave * 16 + 8 * kh;
#pragma unroll
  for (int t = 0; t < NT; ++t) {
#pragma unroll
    for (int j = 0; j < 8; ++j) {
      H[(size_t)(orow + j) * N + t * 16 + m] = acc[t][j];
    }
  }
}

// ---------------- self-loop init (+ optional bias) ----------------
template <int F, bool BIAS>
__global__ void k_selfinit(const float* __restrict__ H, const float* __restrict__ dinv,
                           const float* __restrict__ bias, float* __restrict__ O, int total) {
  int i = blockIdx.x * blockDim.x + threadIdx.x;
  if (i >= total) return;
  int node = i / F;
  int f = i & (F - 1);
  float di = dinv[node];
  float v = H[i] * di * di;
  if (BIAS) v += bias[f];
  O[i] = v;
}

// ---------------- bias + ReLU ----------------
template <int F>
__global__ void k_bias_relu(float* __restrict__ O, const float* __restrict__ bias, int total) {
  int i = blockIdx.x * blockDim.x + threadIdx.x;
  if (i >= total) return;
  O[i] = fmaxf(O[i] + bias[i & (F - 1)], 0.0f);
}

// ---------------- edge scatter: one wave per edge ----------------
template <int FPL>   // floats per lane; F = 32*FPL features
__global__ void k_edge_scatter(const float* __restrict__ H, const float* __restrict__ dinv,
                               const int* __restrict__ rows, const int* __restrict__ cols,
                               float* __restrict__ O, int E) {
  constexpr int F = 32 * FPL;
  int wave = (int)((blockIdx.x * blockDim.x + threadIdx.x) >> 5);
  if (wave >= E) return;                // wave-uniform
  int lane = threadIdx.x & 31;
  int r = rows[wave];
  int c = cols[wave];
  float w = dinv[r] * dinv[c];
  const float* hs = H + (size_t)r * F + lane * FPL;
  float* os = O + (size_t)c * F + lane * FPL;
  float v[FPL];
#pragma unroll
  for (int i = 0; i < FPL; ++i) v[i] = hs[i];   // contiguous -> b64/b128 load
#pragma unroll
  for (int i = 0; i < FPL; ++i) atomAddF(os + i, v[i] * w);
}

// ---------------- launch ----------------
extern "C" void kernel_launch(void* const* d_in, const int* in_sizes, int n_in,
                              void* d_out, int out_size, void* d_ws, size_t ws_size,
                              hipStream_t stream) {
  const float* x  = (const float*)d_in[0];   // [NN, 256]
  const int*   ei = (const int*)d_in[1];     // [2, E]
  const float* W1 = (const float*)d_in[2];   // [256, 128]
  const float* b1 = (const float*)d_in[3];   // [128]
  const float* W2 = (const float*)d_in[4];   // [128, 64]
  const float* b2 = (const float*)d_in[5];   // [64]
  float* out = (float*)d_out;                // [NN, 64]

  const int E = in_sizes[1] / 2;
  const int* rows = ei;
  const int* cols = ei + E;

  // workspace carve-up (floats)
  float* ws   = (float*)d_ws;
  float* dinv = ws;                               // NN (deg -> dinv in place)
  float* h1   = ws + 100032;                      // NN*128
  float* agg1 = h1 + (size_t)NN * 128;            // NN*128
  float* h2   = agg1 + (size_t)NN * 128;          // NN*64

  const int strips = NN / 16;                     // 6250, exact
  const int gemm_blocks = (strips + 7) / 8;       // 8 waves / 256-thread block

  // 1) degrees (self-loop contributes 1.0) -> dinv
  k_fill<<<(NN + 255) / 256, 256, 0, stream>>>(dinv, 1.0f, NN);
  k_deg_accum<<<(E + 255) / 256, 256, 0, stream>>>(dinv, cols, E);
  k_rsqrt_inplace<<<(NN + 255) / 256, 256, 0, stream>>>(dinv, NN);

  // 2) layer 1: h1 = x @ W1   (fp32 WMMA)
  k_gemm_wmma<256, 128, 8><<<gemm_blocks, 256, 0, stream>>>(x, W1, h1, strips);

  // 3) aggregate: agg1 = sum_{(r,c)} h1[r]*dinv[r]*dinv[c]  (+ self loops)
  {
    int total = NN * 128;
    k_selfinit<128, false><<<total / 256, 256, 0, stream>>>(h1, dinv, nullptr, agg1, total);
    k_edge_scatter<4><<<(E + 7) / 8, 256, 0, stream>>>(h1, dinv, rows, cols, agg1, E);
    k_bias_relu<128><<<total / 256, 256, 0, stream>>>(agg1, b1, total);
  }

  // 4) layer 2: h2 = agg1 @ W2   (fp32 WMMA)
  k_gemm_wmma<128, 64, 4><<<gemm_blocks, 256, 0, stream>>>(agg1, W2, h2, strips);

  // 5) aggregate into d_out, bias folded into self-loop init
  {
    int total = NN * 64;
    k_selfinit<64, true><<<total / 256, 256, 0, stream>>>(h2, dinv, b2, out, total);
    k_edge_scatter<2><<<(E + 7) / 8, 256, 0, stream>>>(h2, dinv, rows, cols, out, E);
  }
}